// Sampler_44109314130528
// MI455X (gfx1250) — compile-verified
//
#include <hip/hip_runtime.h>
#include <hip/hip_bf16.h>
#include <stdint.h>

// ---------------------------------------------------------------------------
// Sampler (penalties + temperature + top-k/top-p + softmax) for MI455X.
// Radix-select (4x8-bit rounds on ordered-uint keys) replaces the per-row
// 128k sort; top-p resolved on the tiny kept-set in LDS.
// Streaming kernels use a double-buffered CDNA5 async LDS DMA pipeline:
//   issue tile i+1 -> s_wait_asynccnt 1 -> consume tile i   (no barriers)
// ---------------------------------------------------------------------------

#define TPB    256
#define TILE   1024         // floats per tile (TPB * 4)
#define ITERS  8            // tiles per block (pipelined)
#define CAP    2048         // max kept candidates held in LDS
#define TINY_L 6.103515625e-05f   // float16 tiny, used as a *logit* value

typedef int v4i __attribute__((ext_vector_type(4)));
typedef __attribute__((address_space(1))) v4i* gv4p;
typedef __attribute__((address_space(3))) v4i* lv4p;

// ---------------- CDNA5 async LDS DMA helpers ------------------------------
__device__ __forceinline__ void async_load_b128(const float* gsrc, float* ldst) {
#if __has_builtin(__builtin_amdgcn_global_load_async_to_lds_b128)
  __builtin_amdgcn_global_load_async_to_lds_b128((gv4p)(gsrc), (lv4p)(ldst), 0, 0);
#else
  ((float4*)ldst)[0] = ((const float4*)gsrc)[0];
#endif
}
__device__ __forceinline__ void wait_async0() {
#if __has_builtin(__builtin_amdgcn_s_wait_asynccnt)
  __builtin_amdgcn_s_wait_asynccnt(0);
#else
  asm volatile("s_wait_asynccnt 0" ::: "memory");
#endif
}
__device__ __forceinline__ void wait_async1() {
#if __has_builtin(__builtin_amdgcn_s_wait_asynccnt)
  __builtin_amdgcn_s_wait_asynccnt(1);
#else
  asm volatile("s_wait_asynccnt 1" ::: "memory");
#endif
}

// ---------------- ordered-uint key (monotonic float<->uint) ----------------
__device__ __forceinline__ unsigned f2k(float f) {
  unsigned u = __float_as_uint(f);
  return (u & 0x80000000u) ? ~u : (u | 0x80000000u);
}
__device__ __forceinline__ float k2f(unsigned k) {
  unsigned u = (k & 0x80000000u) ? (k ^ 0x80000000u) : ~k;
  return __uint_as_float(u);
}

// penalties + temperature
__device__ __forceinline__ float adjust(float x, int pm, int om,
                                        float rep, float pres, float temp) {
  if (pm | om) x = (x > 0.0f) ? (x / rep) : (x * rep);
  x -= pres * (float)om;
  float t = (temp < 1e-5f) ? 1.0f : temp;
  return x / t;
}

// ---------------- kernels --------------------------------------------------
__global__ void k_zero(unsigned* p, size_t n) {
  size_t i = (size_t)blockIdx.x * blockDim.x + threadIdx.x;
  size_t stride = (size_t)gridDim.x * blockDim.x;
  for (; i < n; i += stride) p[i] = 0u;
}

__global__ void k_scatter(const int* __restrict__ toks, unsigned* __restrict__ bits,
                          int B, int L, int V, int Vw) {
  long long i = (long long)blockIdx.x * blockDim.x + threadIdx.x;
  if (i >= (long long)B * L) return;
  int b = (int)(i / L);
  int t = toks[i];
  if (t >= 0 && t < V)
    atomicOr(&bits[(size_t)b * Vw + (t >> 5)], 1u << (t & 31));
}

// pass: adjusted logits -> row max key + 256-bin histogram of top byte
__global__ void k_stats(const float* __restrict__ logits,
                        const unsigned* __restrict__ mp, const unsigned* __restrict__ mo,
                        const float* __restrict__ rep, const float* __restrict__ pres,
                        const float* __restrict__ temp,
                        unsigned* __restrict__ rowmax, unsigned* __restrict__ hist0,
                        int V, int Vw, int NB, int NTB) {
  __shared__ float tile[2][TILE];
  __shared__ unsigned hs[256];
  __shared__ unsigned smax;
  int b = blockIdx.x / NB, cb = blockIdx.x % NB;
  int tid = threadIdx.x;
  if (tid < 256) hs[tid] = 0u;
  if (tid == 0) smax = 0u;
  __syncthreads();

  const float* g = logits + (size_t)b * V;
  int t0 = cb * ITERS;
  int nt = min(ITERS, NTB - t0);
  if (nt <= 0) return;
  float rb = rep[b], pb = pres[b], tpb = temp[b];
  unsigned lmax = 0u;

  auto full  = [&](int it) { return (t0 + it + 1) * TILE <= V; };
  auto issue = [&](int it) {
    async_load_b128(g + (t0 + it) * TILE + tid * 4, &tile[it & 1][tid * 4]);
  };

  if (full(0)) issue(0);
  for (int it = 0; it < nt; ++it) {
    bool nxt = (it + 1 < nt) && full(it + 1);
    if (nxt) issue(it + 1);
    int idx = (t0 + it) * TILE + tid * 4;
    float* buf = &tile[it & 1][tid * 4];
    if (full(it)) {
      if (nxt) wait_async1(); else wait_async0();
    } else {
      for (int j = 0; j < 4; ++j) buf[j] = (idx + j < V) ? g[idx + j] : -3.4e38f;
    }
    unsigned pw = (idx < V) ? mp[(size_t)b * Vw + (idx >> 5)] : 0u;
    unsigned ow = (idx < V) ? mo[(size_t)b * Vw + (idx >> 5)] : 0u;
    int bit0 = idx & 31;
    for (int j = 0; j < 4; ++j) {
      int t = idx + j;
      if (t >= V) continue;
      int pm = (pw >> (bit0 + j)) & 1;
      int om = (ow >> (bit0 + j)) & 1;
      float x = adjust(buf[j], pm, om, rb, pb, tpb);
      unsigned key = f2k(x);
      lmax = max(lmax, key);
      atomicAdd(&hs[key >> 24], 1u);
    }
  }
  atomicMax(&smax, lmax);
  __syncthreads();
  if (tid < 256 && hs[tid]) atomicAdd(&hist0[b * 256 + tid], hs[tid]);
  if (tid == 0) atomicMax(&rowmax[b], smax);
}

// radix-select refinement round: histogram next byte where prefix matches
__global__ void k_hist(const float* __restrict__ logits,
                       const unsigned* __restrict__ mp, const unsigned* __restrict__ mo,
                       const float* __restrict__ rep, const float* __restrict__ pres,
                       const float* __restrict__ temp,
                       const unsigned* __restrict__ prefix, unsigned* __restrict__ histr,
                       int pshift, int bshift, int V, int Vw, int NB, int NTB) {
  __shared__ float tile[2][TILE];
  __shared__ unsigned hs[256];
  int b = blockIdx.x / NB, cb = blockIdx.x % NB;
  int tid = threadIdx.x;
  if (tid < 256) hs[tid] = 0u;
  __syncthreads();

  const float* g = logits + (size_t)b * V;
  int t0 = cb * ITERS;
  int nt = min(ITERS, NTB - t0);
  if (nt <= 0) return;
  unsigned pfx = prefix[b];
  float rb = rep[b], pb = pres[b], tpb = temp[b];

  auto full  = [&](int it) { return (t0 + it + 1) * TILE <= V; };
  auto issue = [&](int it) {
    async_load_b128(g + (t0 + it) * TILE + tid * 4, &tile[it & 1][tid * 4]);
  };

  if (full(0)) issue(0);
  for (int it = 0; it < nt; ++it) {
    bool nxt = (it + 1 < nt) && full(it + 1);
    if (nxt) issue(it + 1);
    int idx = (t0 + it) * TILE + tid * 4;
    float* buf = &tile[it & 1][tid * 4];
    if (full(it)) {
      if (nxt) wait_async1(); else wait_async0();
    } else {
      for (int j = 0; j < 4; ++j) buf[j] = (idx + j < V) ? g[idx + j] : -3.4e38f;
    }
    unsigned pw = (idx < V) ? mp[(size_t)b * Vw + (idx >> 5)] : 0u;
    unsigned ow = (idx < V) ? mo[(size_t)b * Vw + (idx >> 5)] : 0u;
    int bit0 = idx & 31;
    for (int j = 0; j < 4; ++j) {
      int t = idx + j;
      if (t >= V) continue;
      int pm = (pw >> (bit0 + j)) & 1;
      int om = (ow >> (bit0 + j)) & 1;
      unsigned key = f2k(adjust(buf[j], pm, om, rb, pb, tpb));
      if ((key >> pshift) == pfx) atomicAdd(&hs[(key >> bshift) & 255u], 1u);
    }
  }
  __syncthreads();
  if (tid < 256 && hs[tid]) atomicAdd(&histr[b * 256 + tid], hs[tid]);
}

// pick the byte containing the krem-th largest; update prefix/krem; r==3 -> thr
__global__ void k_pick(const unsigned* __restrict__ hist, unsigned* __restrict__ prefix,
                       unsigned* __restrict__ krem, const int* __restrict__ topk,
                       float* __restrict__ thr, int r, int B, int V) {
  int b = blockIdx.x * blockDim.x + threadIdx.x;
  if (b >= B) return;
  unsigned k = (r == 0) ? (unsigned)min(max(topk[b], 1), V) : krem[b];
  const unsigned* h = hist + (size_t)b * 256;
  unsigned acc = 0;
  int sel = 0;
  for (int i = 255; i >= 0; --i) {
    unsigned c = h[i];
    if (acc + c >= k) { sel = i; k -= acc; break; }
    acc += c;
  }
  unsigned np = (r == 0) ? (unsigned)sel : ((prefix[b] << 8) | (unsigned)sel);
  prefix[b] = np;
  krem[b] = k;
  if (r == 3) thr[b] = k2f(np);
}

// gather kept values (>= thr) into LDS, bitonic sort, resolve top-p threshold
__global__ void k_kept(const float* __restrict__ logits,
                       const unsigned* __restrict__ mp, const unsigned* __restrict__ mo,
                       const float* __restrict__ rep, const float* __restrict__ pres,
                       const float* __restrict__ temp,
                       const unsigned* __restrict__ rowmax, const float* __restrict__ thr,
                       const float* __restrict__ topp,
                       float* __restrict__ thrF, float* __restrict__ S2out,
                       float* __restrict__ Mout, int V, int Vw) {
  __shared__ float vals[CAP];
  __shared__ unsigned cnt;
  int b = blockIdx.x, tid = threadIdx.x;
  if (tid == 0) cnt = 0u;
  __syncthreads();

  float th = thr[b];
  float rb = rep[b], pb = pres[b], tpb = temp[b];
  const float* g = logits + (size_t)b * V;
  for (int t = tid; t < V; t += blockDim.x) {
    int pm = (mp[(size_t)b * Vw + (t >> 5)] >> (t & 31)) & 1;
    int om = (mo[(size_t)b * Vw + (t >> 5)] >> (t & 31)) & 1;
    float x = adjust(g[t], pm, om, rb, pb, tpb);
    if (x >= th) {
      unsigned p = atomicAdd(&cnt, 1u);
      if (p < CAP) vals[p] = x;
    }
  }
  __syncthreads();
  unsigned Ktot = cnt;
  int N = (int)min(Ktot, (unsigned)CAP);
  int P = 1;
  while (P < N) P <<= 1;
  for (int i = N + tid; i < P; i += blockDim.x) vals[i] = 3.4e38f;
  __syncthreads();

  for (unsigned ks = 2; ks <= (unsigned)P; ks <<= 1) {
    for (unsigned j = ks >> 1; j > 0; j >>= 1) {
      for (unsigned i = tid; i < (unsigned)P; i += blockDim.x) {
        unsigned ij = i ^ j;
        if (ij > i) {
          bool up = ((i & ks) == 0);
          float a = vals[i], c = vals[ij];
          if ((a > c) == up) { vals[i] = c; vals[ij] = a; }
        }
      }
      __syncthreads();
    }
  }

  if (tid == 0) {
    float M = k2f(rowmax[b]);
    float base = (float)(V - (int)Ktot) * __expf(TINY_L - M);
    float S = base;
    for (int i = 0; i < N; ++i) S += __expf(vals[i] - M);
    float lim = 1.0f - topp[b];
    float run = base / S;
    int i0 = N - 1;  // last (== row max) is always kept
    for (int i = 0; i < N; ++i) {
      run += __expf(vals[i] - M) / S;
      if (run > lim) { i0 = i; break; }
    }
    float tf = vals[i0];
    int Kfin = N - i0;
    float S2 = (float)(V - Kfin) * __expf(TINY_L - M);
    for (int i = i0; i < N; ++i) S2 += __expf(vals[i] - M);
    thrF[b] = tf;
    S2out[b] = S2;
    Mout[b] = M;
  }
}

// final pass: probs + both masks
__global__ void k_write(const float* __restrict__ logits,
                        const unsigned* __restrict__ mp, const unsigned* __restrict__ mo,
                        const float* __restrict__ rep, const float* __restrict__ pres,
                        const float* __restrict__ temp,
                        const float* __restrict__ thrF, const float* __restrict__ S2,
                        const float* __restrict__ Mv,
                        float* __restrict__ out, int B, int V, int Vw, int NB, int NTB) {
  __shared__ float tile[2][TILE];
  int b = blockIdx.x / NB, cb = blockIdx.x % NB;
  int tid = threadIdx.x;

  const float* g = logits + (size_t)b * V;
  int t0 = cb * ITERS;
  int nt = min(ITERS, NTB - t0);
  if (nt <= 0) return;
  float rb = rep[b], pb = pres[b], tpb = temp[b];
  float tf = thrF[b], s2 = S2[b], m = Mv[b];
  size_t BV = (size_t)B * V;

  auto full  = [&](int it) { return (t0 + it + 1) * TILE <= V; };
  auto issue = [&](int it) {
    async_load_b128(g + (t0 + it) * TILE + tid * 4, &tile[it & 1][tid * 4]);
  };

  if (full(0)) issue(0);
  for (int it = 0; it < nt; ++it) {
    bool nxt = (it + 1 < nt) && full(it + 1);
    if (nxt) issue(it + 1);
    int idx = (t0 + it) * TILE + tid * 4;
    float* buf = &tile[it & 1][tid * 4];
    if (full(it)) {
      if (nxt) wait_async1(); else wait_async0();
    } else {
      for (int j = 0; j < 4; ++j) buf[j] = (idx + j < V) ? g[idx + j] : -3.4e38f;
    }
    unsigned pw = (idx < V) ? mp[(size_t)b * Vw + (idx >> 5)] : 0u;
    unsigned ow = (idx < V) ? mo[(size_t)b * Vw + (idx >> 5)] : 0u;
    int bit0 = idx & 31;
    for (int j = 0; j < 4; ++j) {
      int t = idx + j;
      if (t >= V) continue;
      int pm = (pw >> (bit0 + j)) & 1;
      int om = (ow >> (bit0 + j)) & 1;
      float x = adjust(buf[j], pm, om, rb, pb, tpb);
      float w = (x >= tf) ? x : TINY_L;
      size_t o = (size_t)b * V + t;
      out[o] = __expf(w - m) / s2;
      out[BV + o] = pm ? 1.0f : 0.0f;
      out[2 * BV + o] = om ? 1.0f : 0.0f;
    }
  }
}

// ---------------- host launch ---------------------------------------------
extern "C" void kernel_launch(void* const* d_in, const int* in_sizes, int n_in,
                              void* d_out, int out_size, void* d_ws, size_t ws_size,
                              hipStream_t stream) {
  const float* logits = (const float*)d_in[0];
  const int* prompt = (const int*)d_in[1];
  const int* output = (const int*)d_in[2];
  const float* pres = (const float*)d_in[3];
  // d_in[4] = frequency_penalties: computed-but-unused in the reference
  const float* rep = (const float*)d_in[5];
  const float* temp = (const float*)d_in[6];
  const int* topk = (const int*)d_in[7];
  const float* topp = (const float*)d_in[8];

  int B = in_sizes[3];
  int V = in_sizes[0] / B;
  int PL = in_sizes[1] / B;
  int OL = in_sizes[2] / B;
  int Vw = (V + 31) / 32;
  int NTB = (V + TILE - 1) / TILE;       // tiles per row
  int NB = (NTB + ITERS - 1) / ITERS;    // blocks per row (pipelined tiles)

  // workspace layout (all contiguous; zeroed region comes first)
  char* w = (char*)d_ws;
  unsigned* maskP = (unsigned*)w;            w += (size_t)B * Vw * 4;
  unsigned* maskO = (unsigned*)w;            w += (size_t)B * Vw * 4;
  unsigned* rowmax = (unsigned*)w;           w += (size_t)B * 4;
  unsigned* hist = (unsigned*)w;             w += (size_t)4 * B * 256 * 4;  // 4 rounds
  unsigned* prefix = (unsigned*)w;           w += (size_t)B * 4;
  unsigned* krem = (unsigned*)w;             w += (size_t)B * 4;
  float* thr = (float*)w;                    w += (size_t)B * 4;
  float* thrF = (float*)w;                   w += (size_t)B * 4;
  float* S2 = (float*)w;                     w += (size_t)B * 4;
  float* Mv = (float*)w;                     w += (size_t)B * 4;

  size_t nzero = (size_t)2 * B * Vw + B + (size_t)4 * B * 256;
  k_zero<<<1024, TPB, 0, stream>>>(maskP, nzero);

  {
    long long n = (long long)B * PL;
    k_scatter<<<(unsigned)((n + TPB - 1) / TPB), TPB, 0, stream>>>(prompt, maskP, B, PL, V, Vw);
  }
  {
    long long n = (long long)B * OL;
    k_scatter<<<(unsigned)((n + TPB - 1) / TPB), TPB, 0, stream>>>(output, maskO, B, OL, V, Vw);
  }

  dim3 gridBV(B * NB);
  k_stats<<<gridBV, TPB, 0, stream>>>(logits, maskP, maskO, rep, pres, temp,
                                      rowmax, hist, V, Vw, NB, NTB);
  k_pick<<<(B + 127) / 128, 128, 0, stream>>>(hist, prefix, krem, topk, thr, 0, B, V);

  for (int r = 1; r <= 3; ++r) {
    unsigned* hr = hist + (size_t)r * B * 256;
    k_hist<<<gridBV, TPB, 0, stream>>>(logits, maskP, maskO, rep, pres, temp,
                                       prefix, hr, 32 - 8 * r, 24 - 8 * r, V, Vw, NB, NTB);
    k_pick<<<(B + 127) / 128, 128, 0, stream>>>(hr, prefix, krem, topk, thr, r, B, V);
  }

  k_kept<<<B, TPB, 0, stream>>>(logits, maskP, maskO, rep, pres, temp,
                                rowmax, thr, topp, thrF, S2, Mv, V, Vw);

  k_write<<<gridBV, TPB, 0, stream>>>(logits, maskP, maskO, rep, pres, temp,
                                      thrF, S2, Mv, (float*)d_out, B, V, Vw, NB, NTB);
}